// DGLayer_16286515986763
// MI455X (gfx1250) — compile-verified
//
#include <hip/hip_runtime.h>
#include <math.h>

// ---------------------------------------------------------------------------
// DGLayer forward for MI455X (gfx1250, wave32).
// One workgroup (256 thr = 8 waves) per row of 1024. Memory-bound (~384MB
// total traffic -> ~16.5us roofline at 23.3 TB/s).
// Row-sum reductions routed through V_WMMA_F32_16X16X4_F32 (A = ones trick).
// Exact top-k=32 threshold via 4-pass 256-bin radix select on float bits.
// cos() uses the native v_cos_f32 path (__cosf): phase is in [0, 2pi) by
// construction, so the libm Payne-Hanek reduction is pure VALU overhead.
// ---------------------------------------------------------------------------

typedef float v2f __attribute__((ext_vector_type(2)));
typedef float v8f __attribute__((ext_vector_type(8)));

#define N_INPUT   1024
#define TOPK      32u
#define THREADS   256

// Block-wide sum of 256 per-thread partials using the matrix pipe.
// With A = all-ones (16x4), D[m][n] = sum_k B[k][n]: every row of D holds the
// column sums, so summing one D-row across its 16 lanes yields the total of
// all 64 B elements regardless of the exact lane<->element mapping. 4
// accumulating WMMAs cover all 256 partials; all 8 waves do it redundantly so
// EXEC is all ones everywhere (WMMA requirement) and no broadcast round-trip
// is needed: after the xor-reduce, lanes 0-15 hold sum(D[0][*]) and lanes
// 16-31 hold sum(D[8][*]) -- both equal the grand total.
__device__ __forceinline__ float block_sum_wmma(float v, float* red, int tid) {
  red[tid] = v;
  __syncthreads();
  const int lane = tid & 31;
  v2f a; a[0] = 1.0f; a[1] = 1.0f;
  v8f c = {};
#pragma unroll
  for (int chunk = 0; chunk < 4; ++chunk) {
    v2f b;
    b[0] = red[chunk * 64 + lane];
    b[1] = red[chunk * 64 + 32 + lane];
    // 8 args: (neg_a, A, neg_b, B, c_mod, C, reuse_a, reuse_b)
    c = __builtin_amdgcn_wmma_f32_16x16x4_f32(false, a, false, b,
                                              (short)0, c, false, false);
  }
  float s = c[0];
  s += __shfl_xor(s, 1, 32);
  s += __shfl_xor(s, 2, 32);
  s += __shfl_xor(s, 4, 32);
  s += __shfl_xor(s, 8, 32);
  __syncthreads();   // red[] is reused by the caller
  return s;
}

__global__ __launch_bounds__(THREADS)
void dg_layer_kernel(const float* __restrict__ phase,
                     const float* __restrict__ amp,
                     const float* __restrict__ ffi_scale_p,
                     float* __restrict__ out) {
  __shared__ float    red[THREADS];
  __shared__ unsigned hist[256];
  __shared__ unsigned bcast[2];

  const int tid = threadIdx.x;
  const long long row = blockIdx.x;
  const float4* ph4 = (const float4*)(phase + row * N_INPUT);
  const float4* am4 = (const float4*)(amp   + row * N_INPUT);
  const float4 ph = ph4[tid];
  const float4 am = am4[tid];
  const float scale = fmaxf(ffi_scale_p[0], 0.01f);   // clip(ffi_scale, 0.01, None)

  // drive = amp*scale * 0.5*(1+cos(phase)), all >= 0.
  // __cosf -> v_cos_f32 (native); phase is uniform in [0, 2pi).
  float d[4];
  d[0] = am.x * scale * 0.5f * (1.0f + __cosf(ph.x));
  d[1] = am.y * scale * 0.5f * (1.0f + __cosf(ph.y));
  d[2] = am.z * scale * 0.5f * (1.0f + __cosf(ph.z));
  d[3] = am.w * scale * 0.5f * (1.0f + __cosf(ph.w));

  // Reduction #1: mean of drive -> m0, m1 (steps 0,1 collapse analytically:
  // ffi=0 and drive>=0 make relu a no-op, so ema1 = 0.51*drive exactly).
  const float mean_d = block_sum_wmma(d[0] + d[1] + d[2] + d[3], red, tid)
                       * (1.0f / (float)N_INPUT);
  const float m0 = 0.3f  * mean_d;   // mean(ema after t=0)
  const float m1 = 0.51f * mean_d;   // mean(ema after t=1)

  // ema after t=2: 0.7*0.51*d + 0.3*relu(d-m0)
  float e2[4];
#pragma unroll
  for (int j = 0; j < 4; ++j)
    e2[j] = 0.357f * d[j] + 0.3f * fmaxf(d[j] - m0, 0.0f);

  // Reduction #2: m2 = mean(ema after t=2)
  const float m2 = block_sum_wmma(e2[0] + e2[1] + e2[2] + e2[3], red, tid)
                   * (1.0f / (float)N_INPUT);

  // t=3 uses m1, t=4 uses m2; m3/m4 are never consumed (FBI_DELAY=20 > 5
  // also means the per-step WTA masks are dead code in the reference).
  float e4[4];
  unsigned u[4];
#pragma unroll
  for (int j = 0; j < 4; ++j) {
    const float e3 = 0.7f * e2[j] + 0.3f * fmaxf(d[j] - m1, 0.0f);
    e4[j]          = 0.7f * e3    + 0.3f * fmaxf(d[j] - m2, 0.0f);
    u[j] = __float_as_uint(e4[j]);   // nonneg floats order like their bits
  }

  // ---- exact kth-largest (k=32) via MSB-first radix select, 4x 8-bit passes
  unsigned prefix = 0;
  unsigned rk = TOPK;
#pragma unroll 1
  for (int p = 0; p < 4; ++p) {
    const int shift = 24 - 8 * p;
    hist[tid] = 0;
    __syncthreads();
#pragma unroll
    for (int j = 0; j < 4; ++j) {
      const unsigned uu = u[j];
      const int s8 = shift + 8;
      const bool cand = (s8 >= 32) || ((uu >> s8) == prefix);
      if (cand) atomicAdd(&hist[(uu >> shift) & 0xFFu], 1u);
    }
    __syncthreads();
    if (tid < 32) {                     // wave 0: pick the crossing bin
      const int lane = tid;
      const int dhi  = 255 - 8 * lane;  // lane covers digits dhi..dhi-7 (desc)
      unsigned cv[8];
      unsigned csum = 0;
#pragma unroll
      for (int i = 0; i < 8; ++i) { cv[i] = hist[dhi - i]; csum += cv[i]; }
      unsigned incl = csum;             // inclusive scan over descending lanes
#pragma unroll
      for (int off = 1; off < 32; off <<= 1) {
        const unsigned t = __shfl_up(incl, off, 32);
        if (lane >= off) incl += t;
      }
      const unsigned excl = incl - csum;
      if (excl < rk && incl >= rk) {    // exactly one lane hits
        unsigned run = excl, dsel = 0, nrk = rk;
#pragma unroll
        for (int i = 0; i < 8; ++i) {
          if (run + cv[i] >= rk) { dsel = (unsigned)(dhi - i); nrk = rk - run; break; }
          run += cv[i];
        }
        bcast[0] = dsel;
        bcast[1] = nrk;
      }
    }
    __syncthreads();
    prefix = (prefix << 8) | bcast[0];
    rk = bcast[1];
    __syncthreads();
  }
  const unsigned T = prefix;            // bit pattern of kth-largest value

  // final WTA: keep ema >= thresh (uint compare == float compare for >=0)
  float4 o;
  o.x = (u[0] >= T) ? e4[0] : 0.0f;
  o.y = (u[1] >= T) ? e4[1] : 0.0f;
  o.z = (u[2] >= T) ? e4[2] : 0.0f;
  o.w = (u[3] >= T) ? e4[3] : 0.0f;
  ((float4*)(out + row * N_INPUT))[tid] = o;
}

extern "C" void kernel_launch(void* const* d_in, const int* in_sizes, int n_in,
                              void* d_out, int out_size, void* d_ws, size_t ws_size,
                              hipStream_t stream) {
  const float* phase = (const float*)d_in[0];
  const float* amp   = (const float*)d_in[1];
  const float* scale = (const float*)d_in[2];
  // d_in[3] (fbi_temperature) is unused in the reference forward.
  float* out = (float*)d_out;

  const int B = in_sizes[0] / N_INPUT;   // 32768 rows
  dg_layer_kernel<<<B, THREADS, 0, stream>>>(phase, amp, scale, out);
}